// GssLogNormLayer_11974368821394
// MI455X (gfx1250) — compile-verified
//
#include <hip/hip_runtime.h>

// GssLogNorm: per-element golden-section search maximizing
//   surplus(d) = bid * (1-d) * 0.5*(1+erf((ln(d*bid)-mu)/(sigma*sqrt2)))
// over d in [0,1], 20 iterations, output midpoint of final bracket.
//
// TRANS/VALU-bound: minimize transcendental count.
//   z(d) = (log2(d) + c0) * scale,  c0 = log2(bid) - mu/ln2,
//   scale = (ln2/sqrt2) * rcp(sigma)
// Comparison drops the common positive factors bid and 0.5.
// GSS probe-reuse: with ratio phi, the surviving probe of iteration t is one
// of the probes of iteration t+1, so only ONE new eval (1 v_log + 1 v_rcp +
// 1 v_exp) is needed per iteration instead of two.

#define INV_GR      0.61803398874989484820f   // 1/phi
#define INV_LN2     1.4426950408889634f       // 1/ln(2)
#define LN2_OSQRT2  0.49012907173427387f      // ln(2)/sqrt(2)
#define LOG2E       1.4426950408889634f

__device__ __forceinline__ float erf_fast(float x) {
    // Abramowitz & Stegun 7.1.26, |err| < 1.5e-7. One v_exp, one v_rcp, 6 FMA.
    float a = __builtin_fabsf(x);
    float t = __builtin_amdgcn_rcpf(__builtin_fmaf(0.3275911f, a, 1.0f));
    float p = 1.061405429f;
    p = __builtin_fmaf(p, t, -1.453152027f);
    p = __builtin_fmaf(p, t,  1.421413741f);
    p = __builtin_fmaf(p, t, -0.284496736f);
    p = __builtin_fmaf(p, t,  0.254829592f);
    float e = __builtin_amdgcn_exp2f(-(a * a) * LOG2E);   // v_exp_f32
    float r = __builtin_fmaf(-(p * t), e, 1.0f);          // r in [0,1)
    return __builtin_copysignf(r, x);
}

// Relative surplus at probe d (common positive factors bid, 0.5 dropped):
//   (1-d) * (1 + erf((log2(d) + c0) * scale))
__device__ __forceinline__ float surplus_rel(float d, float c0, float scale) {
    float z = (__builtin_amdgcn_logf(d) + c0) * scale;    // v_log_f32
    return (1.0f - d) * (1.0f + erf_fast(z));
}

__global__ __launch_bounds__(256) void gss_lognorm_kernel(
    const float* __restrict__ params,     // [N,2] interleaved (mu, sigma)
    const float* __restrict__ bid_prices, // [N]
    float* __restrict__ out,              // [N]
    int n4)                               // N/4 (N is a multiple of 4)
{
    const int i = blockIdx.x * blockDim.x + threadIdx.x;
    if (i >= n4) return;

    // 128-bit vector loads: 4 bids + 4 (mu,sigma) pairs.
    const float4 b  = reinterpret_cast<const float4*>(bid_prices)[i];
    const float4 p0 = reinterpret_cast<const float4*>(params)[2 * i + 0]; // mu0,s0,mu1,s1
    const float4 p1 = reinterpret_cast<const float4*>(params)[2 * i + 1]; // mu2,s2,mu3,s3

    const float bv[4] = { b.x,  b.y,  b.z,  b.w  };
    const float mu[4] = { p0.x, p0.z, p1.x, p1.z };
    const float sg[4] = { p0.y, p0.w, p1.y, p1.w };

    float c0[4], scale[4], bmin[4], bmax[4];
    float x1[4], x2[4], f1[4], f2[4];
#pragma unroll
    for (int k = 0; k < 4; ++k) {
        c0[k]    = __builtin_fmaf(-mu[k], INV_LN2, __builtin_amdgcn_logf(bv[k]));
        scale[k] = __builtin_amdgcn_rcpf(sg[k]) * LN2_OSQRT2;
        bmin[k]  = 0.0f;                 // FLOOR
        bmax[k]  = 1.0f;
        x1[k]    = 1.0f - INV_GR;
        x2[k]    = INV_GR;
        f1[k]    = surplus_rel(x1[k], c0[k], scale[k]);
        f2[k]    = surplus_rel(x2[k], c0[k], scale[k]);
    }

    // 19 full iterations with probe reuse (one eval each); 4 independent
    // chains interleaved to hide v_log/v_exp/v_rcp latency. Outer loop kept
    // rolled so the hot body stays I-cache resident.
#pragma unroll 1
    for (int it = 0; it < 19; ++it) {
#pragma unroll
        for (int k = 0; k < 4; ++k) {
            const bool cond = f1[k] > f2[k];
            bmax[k] = cond ? x2[k]   : bmax[k];
            bmin[k] = cond ? bmin[k] : x1[k];

            const float xold = cond ? x1[k] : x2[k];   // surviving probe
            const float fold = cond ? f1[k] : f2[k];

            const float w    = (bmax[k] - bmin[k]) * INV_GR;
            const float xnew = cond ? (bmax[k] - w) : (bmin[k] + w);
            const float fnew = surplus_rel(xnew, c0[k], scale[k]);

            x1[k] = cond ? xnew : xold;
            f1[k] = cond ? fnew : fold;
            x2[k] = cond ? xold : xnew;
            f2[k] = cond ? fold : fnew;
        }
    }

    // 20th iteration: bracket update only (its probe evals would be unused).
    float r[4];
#pragma unroll
    for (int k = 0; k < 4; ++k) {
        const bool cond = f1[k] > f2[k];
        bmax[k] = cond ? x2[k]   : bmax[k];
        bmin[k] = cond ? bmin[k] : x1[k];
        r[k] = (bmin[k] + bmax[k]) * 0.5f;
    }

    reinterpret_cast<float4*>(out)[i] = make_float4(r[0], r[1], r[2], r[3]);
}

extern "C" void kernel_launch(void* const* d_in, const int* in_sizes, int n_in,
                              void* d_out, int out_size, void* d_ws, size_t ws_size,
                              hipStream_t stream) {
    const float* params = (const float*)d_in[0];      // [N,2]
    const float* bids   = (const float*)d_in[1];      // [N]
    float*       out    = (float*)d_out;              // [N]

    const int n  = in_sizes[1];   // N = 16777216 (multiple of 4)
    const int n4 = n >> 2;

    const int block = 256;                  // 8 wave32 waves per block
    const int grid  = (n4 + block - 1) / block;
    gss_lognorm_kernel<<<grid, block, 0, stream>>>(params, bids, out, n4);
}